// CKT_dev_6828998001204
// MI455X (gfx1250) — compile-verified
//
#include <hip/hip_runtime.h>
#include <math.h>

// Problem constants (match reference)
constexpr int B  = 32;
constexpr int T  = 512;
constexpr int K  = 110;
constexpr int C  = 20;
constexpr int EK = 100;   // knowledge emb
constexpr int EI = 200;   // interaction emb

typedef float v2f __attribute__((ext_vector_type(2)));
typedef float v8f __attribute__((ext_vector_type(8)));

// ---------------------------------------------------------------------------
// Kernel 1: logits via V_WMMA_F32_16X16X4_F32 + softmax -> w[B*T, C]
// One wave (32 threads) per block; each block owns 16 (b,t) rows.
// ---------------------------------------------------------------------------
__global__ __launch_bounds__(32) void ckt_logits_softmax(
    const int* __restrict__ x,              // [B*T]
    const float* __restrict__ question_emb, // [K+1, EK]
    const float* __restrict__ key_memory,   // [C, EK]
    float* __restrict__ w_out)              // [B*T, C]
{
    __shared__ float sQ[16][EK];   // 16 q rows
    __shared__ float sK[32][EK];   // key padded to 32 rows (rows >= C are zero)
    __shared__ float sL[16][32];   // logits tile

    const int lane    = threadIdx.x;        // 0..31
    const int rowbase = blockIdx.x * 16;    // base (b*T+t) row

    // Stage zero-padded key matrix: 32*EK floats, 3200 % 32 == 0 -> no divergence
    for (int i = lane; i < 32 * EK; i += 32) {
        int c = i / EK, e = i % EK;
        sK[c][e] = (c < C) ? key_memory[c * EK + e] : 0.0f;
    }
    // Stage 16 question-embedding rows: 1600 % 32 == 0 -> no divergence
    for (int i = lane; i < 16 * EK; i += 32) {
        int r = i / EK, e = i % EK;
        int xv  = x[rowbase + r];
        int qid = ((xv - 1) % K + K) % K + 1;   // Python-style modulo
        sQ[r][e] = question_emb[qid * EK + e];
    }
    __syncthreads();

    // WMMA f32 16x16x4 lane layouts (ISA 7.12.2):
    //  A (16x4): lane m = lane&15 holds row M=m; VGPR0 K = (lane<16?0:2), VGPR1 K+1
    //  B (4x16): lane n = lane&15 holds col N=n; VGPR0 K = (lane<16?0:2), VGPR1 K+1
    //  D (16x16): VGPR v holds row v + (lane<16?0:8), col lane&15
    const int m    = lane & 15;
    const int koff = (lane < 16) ? 0 : 2;

    v8f acc0 = {};  // columns c = 0..15
    v8f acc1 = {};  // columns c = 16..31 (only 16..19 meaningful)

    #pragma unroll
    for (int k0 = 0; k0 < EK; k0 += 4) {    // 25 K-steps
        v2f a, b0, b1;
        a.x  = sQ[m][k0 + koff];
        a.y  = sQ[m][k0 + koff + 1];
        b0.x = sK[m][k0 + koff];
        b0.y = sK[m][k0 + koff + 1];
        b1.x = sK[16 + m][k0 + koff];
        b1.y = sK[16 + m][k0 + koff + 1];
        acc0 = __builtin_amdgcn_wmma_f32_16x16x4_f32(
                   false, a, false, b0, (short)0, acc0, false, false);
        acc1 = __builtin_amdgcn_wmma_f32_16x16x4_f32(
                   false, a, false, b1, (short)0, acc1, false, false);
    }

    // Spill the 16x32 logits tile to LDS per the D layout.
    const int rbase = (lane < 16) ? 0 : 8;
    #pragma unroll
    for (int v = 0; v < 8; v++) {
        sL[rbase + v][m]      = acc0[v];
        sL[rbase + v][16 + m] = acc1[v];
    }
    __syncthreads();

    // Softmax over c = 0..C-1; lanes 0..15 each own one row.
    if (lane < 16) {
        float mx = -INFINITY;
        #pragma unroll
        for (int c = 0; c < C; c++) mx = fmaxf(mx, sL[lane][c]);
        float ev[C];
        float s = 0.0f;
        #pragma unroll
        for (int c = 0; c < C; c++) { ev[c] = __expf(sL[lane][c] - mx); s += ev[c]; }
        float inv = 1.0f / s;
        int bt = rowbase + lane;
        #pragma unroll
        for (int c = 0; c < C; c++) w_out[bt * C + c] = ev[c] * inv;
    }
}

// ---------------------------------------------------------------------------
// Kernel 2: inter[B*T, EI] = tanh(interaction_emb[x])
// ---------------------------------------------------------------------------
__global__ __launch_bounds__(256) void ckt_inter(
    const int* __restrict__ x,                 // [B*T]
    const float* __restrict__ interaction_emb, // [2K+1, EI]
    float* __restrict__ inter_ws)              // [B*T, EI]
{
    int idx = blockIdx.x * blockDim.x + threadIdx.x;
    if (idx < B * T * EI) {
        int bt = idx / EI;
        int e  = idx % EI;
        inter_ws[idx] = tanhf(interaction_emb[x[bt] * EI + e]);
    }
}

// ---------------------------------------------------------------------------
// Kernel 3: serial-over-T prefix scan of rank-1 updates (the HBM-bound pass).
// One block per (b,c); thread e owns one EI lane -> coalesced 800B stores / t.
// ---------------------------------------------------------------------------
__global__ __launch_bounds__(256) void ckt_scan(
    const float* __restrict__ w_ws,     // [B*T, C]
    const float* __restrict__ inter_ws, // [B*T, EI]
    const float* __restrict__ vm_init,  // [B, C, EI]
    float* __restrict__ out)            // [B, T, C, EI]
{
    const int e = threadIdx.x;
    if (e >= EI) return;
    const int b = blockIdx.x / C;
    const int c = blockIdx.x % C;

    float acc = vm_init[(b * C + c) * EI + e];

    const float* wp = w_ws    + b * T * C  + c;   // stride C over t (uniform per block)
    const float* ip = inter_ws + b * T * EI + e;  // stride EI over t (coalesced)
    float*       op = out + (b * T * C + c) * EI + e; // stride C*EI over t

    for (int t = 0; t < T; t++) {
        // prefetch interaction rows a few iterations ahead (global_prefetch_b8)
        if (t + 8 < T) __builtin_prefetch(ip + (t + 8) * EI, 0, 0);
        acc = fmaf(wp[t * C], ip[t * EI], acc);
        op[t * C * EI] = acc;
    }
}

// ---------------------------------------------------------------------------
extern "C" void kernel_launch(void* const* d_in, const int* in_sizes, int n_in,
                              void* d_out, int out_size, void* d_ws, size_t ws_size,
                              hipStream_t stream) {
    const int*   x               = (const int*)  d_in[0]; // [B,T]
    // d_in[1] = next_question (unused by the reference output)
    const float* question_emb    = (const float*)d_in[2]; // [K+1,EK]
    const float* interaction_emb = (const float*)d_in[3]; // [2K+1,EI]
    const float* key_memory      = (const float*)d_in[4]; // [C,EK]
    const float* vm_init         = (const float*)d_in[5]; // [B,C,EI]
    float*       out             = (float*)d_out;         // [B,T,C,EI]

    // Workspace layout: w[B*T*C] then inter[B*T*EI]
    float* w_ws     = (float*)d_ws;
    float* inter_ws = w_ws + (size_t)B * T * C;

    // Phase 1: WMMA logits + softmax -> w   (1024 blocks x 1 wave)
    ckt_logits_softmax<<<(B * T) / 16, 32, 0, stream>>>(x, question_emb, key_memory, w_ws);

    // Phase 2: tanh(interaction gather) -> inter
    ckt_inter<<<(B * T * EI + 255) / 256, 256, 0, stream>>>(x, interaction_emb, inter_ws);

    // Phase 3: HBM-bound prefix scan -> out (640 blocks x 256 threads)
    ckt_scan<<<B * C, 256, 0, stream>>>(w_ws, inter_ws, vm_init, out);
}